// Generator_78228534329939
// MI455X (gfx1250) — compile-verified
//
#include <hip/hip_runtime.h>
#include <cstdint>
#include <cstddef>

#define LUT_D 8
#define LUT_H 256
#define LUT_W 256
#define IMG_H 1024
#define IMG_W 1024
#define NBATCH 4
#define BLOCK 256
#define NCA 68     // staged LUT columns (even-aligned window, 64 main + 4 edge)
#define STR 68     // LDS row stride in floats (even -> b64-aligned rows)
#define NROWS 72   // 9 (c,i) volumes * 8 z-slices

typedef int v2i __attribute__((ext_vector_type(2)));

#if defined(__has_builtin)
#if __has_builtin(__builtin_amdgcn_global_load_async_to_lds_b64)
#define HAVE_ASYNC_B64 1
#endif
#if __has_builtin(__builtin_amdgcn_global_load_async_to_lds_b32)
#define HAVE_ASYNC_B32 1
#endif
#endif
#if defined(HAVE_ASYNC_B64) || defined(HAVE_ASYNC_B32)
#define HAVE_ASYNC_LDS 1
#endif

#if defined(__has_builtin)
#if __has_builtin(__builtin_amdgcn_tanhf)
#define TANHF(v) __builtin_amdgcn_tanhf(v)   // v_tanh_f32 (gfx1250 TRANS op)
#endif
#endif
#ifndef TANHF
#define TANHF(v) tanhf(v)
#endif

typedef __attribute__((address_space(1))) int* g32_t;   // global int*
typedef __attribute__((address_space(3))) int* l32_t;   // LDS int*
typedef __attribute__((address_space(1))) v2i* g64_t;   // global int2*
typedef __attribute__((address_space(3))) v2i* l64_t;   // LDS int2*

__global__ __launch_bounds__(BLOCK) void lut3d_tanh_kernel(
    const float* __restrict__ x, const float* __restrict__ param,
    float* __restrict__ out) {
  // Pre-lerped (over y) LUT window: S[ch][col], ch = c*24 + i*8 + z
  __shared__ __attribute__((aligned(16))) float S[NROWS * STR];
#if defined(HAVE_ASYNC_LDS)
  // Raw rows: [ch*2 + {y0,y1}][col] ; 16B-aligned so b64/float2 ops are legal.
  __shared__ __attribute__((aligned(16))) float R[2 * NROWS * STR];
#endif

  const int tid = threadIdx.x;
  const int blk = blockIdx.x;
  const int wblocks = IMG_W / BLOCK;           // 4
  const int w0 = (blk % wblocks) * BLOCK;
  const int h  = (blk / wblocks) % IMG_H;
  const int b  = blk / (wblocks * IMG_H);

  const int w = w0 + tid;
  const size_t cstride = (size_t)IMG_H * IMG_W;
  const size_t pix = (size_t)b * 3 * cstride + (size_t)h * IMG_W + w;

  __builtin_prefetch(&x[pix], 0, 0);
  __builtin_prefetch(&x[pix + cstride], 0, 0);
  __builtin_prefetch(&x[pix + 2 * cstride], 0, 0);

  // ---- block-uniform y coordinate ----
  const float sy = 2.0f / (float)(IMG_H - 1);
  float gy = fmaf((float)h, sy, -1.0f);
  float iy = (gy + 1.0f) * 0.5f * (float)(LUT_H - 1);
  iy = fminf(fmaxf(iy, 0.0f), (float)(LUT_H - 1));
  float y0f = floorf(iy);
  int y0 = (int)y0f;
  int y1 = min(y0 + 1, LUT_H - 1);
  float wy = iy - y0f;

  // ---- block-uniform base LUT column (even-aligned) ----
  const float sx = 2.0f / (float)(IMG_W - 1);
  float gx0 = fmaf((float)w0, sx, -1.0f);
  float ix0 = (gx0 + 1.0f) * 0.5f * (float)(LUT_W - 1);
  ix0 = fminf(fmaxf(ix0, 0.0f), (float)(LUT_W - 1));
  const int x_base = (int)floorf(ix0);   // <= 191
  const int x_al = x_base & ~1;          // even, <= 190

  const float* pb = param + (size_t)b * NROWS * LUT_H * LUT_W;

  // =========== Stage raw LUT rows into LDS (async) ===========
#if defined(HAVE_ASYNC_B64)
  {
    const int cc2 = (tid & 31) * 2;          // 0,2,...,62
    const int rg  = tid >> 5;                // 0..7
    const int ch0 = rg >> 1;                 // 0..3
    const int yy  = (rg & 1) ? y1 : y0;      // fixed per lane
    // x_al + cc2 + 1 <= 190 + 63 = 253 < 255 -> no clamp needed here.
    const float* gp0 = pb + ((size_t)ch0 * LUT_H + (size_t)yy) * LUT_W
                          + (size_t)(x_al + cc2);
    g64_t gsrc = (g64_t)(uintptr_t)gp0;
    l64_t ldst = (l64_t)(uint32_t)(uintptr_t)&R[rg * STR + cc2];
#pragma unroll
    for (int it = 0; it < 18; ++it) {
      __builtin_amdgcn_global_load_async_to_lds_b64(gsrc, ldst, 0, 0);
      gsrc += 2 * LUT_H * LUT_W;             // +4 channels (in int2 units)
      ldst += 4 * STR;                       // +8 raw rows  (in int2 units)
    }
    for (int k = tid; k < 2 * NROWS * 4; k += BLOCK) {  // cols 64..67
      int rr = k >> 2;
      int c4 = 64 + (k & 3);
      int ch = rr >> 1;
      int yy2 = (rr & 1) ? y1 : y0;
      int g = min(x_al + c4, LUT_W - 1);
      const float* gp = pb + ((size_t)ch * LUT_H + (size_t)yy2) * LUT_W + (size_t)g;
      __builtin_amdgcn_global_load_async_to_lds_b32(
          (g32_t)(uintptr_t)gp,
          (l32_t)(uint32_t)(uintptr_t)&R[rr * STR + c4], 0, 0);
    }
  }
#elif defined(HAVE_ASYNC_B32)
  {
    const int cc = tid & 63;
    const int rg = tid >> 6;                 // 0..3
    const int ch0 = rg >> 1;
    const int yy  = (rg & 1) ? y1 : y0;
    const float* gp0 = pb + ((size_t)ch0 * LUT_H + (size_t)yy) * LUT_W
                          + (size_t)(x_al + cc);
    g32_t gsrc = (g32_t)(uintptr_t)gp0;
    l32_t ldst = (l32_t)(uint32_t)(uintptr_t)&R[rg * STR + cc];
#pragma unroll
    for (int it = 0; it < 36; ++it) {
      __builtin_amdgcn_global_load_async_to_lds_b32(gsrc, ldst, 0, 0);
      gsrc += 2 * LUT_H * LUT_W;
      ldst += 4 * STR;
    }
    for (int k = tid; k < 2 * NROWS * 4; k += BLOCK) {
      int rr = k >> 2;
      int c4 = 64 + (k & 3);
      int ch = rr >> 1;
      int yy2 = (rr & 1) ? y1 : y0;
      int g = min(x_al + c4, LUT_W - 1);
      const float* gp = pb + ((size_t)ch * LUT_H + (size_t)yy2) * LUT_W + (size_t)g;
      __builtin_amdgcn_global_load_async_to_lds_b32(
          (g32_t)(uintptr_t)gp,
          (l32_t)(uint32_t)(uintptr_t)&R[rr * STR + c4], 0, 0);
    }
  }
#endif

#if defined(HAVE_ASYNC_LDS)
#if __has_builtin(__builtin_amdgcn_s_wait_asynccnt)
  __builtin_amdgcn_s_wait_asynccnt(0);
#else
  asm volatile("s_wait_asynccnt 0" ::: "memory");
#endif
  __syncthreads();
  {
    // y-lerp pass: float2 on 8B-aligned LDS. S row r <- raw rows 2r, 2r+1.
    const int cc2 = (tid & 31) * 2;
    const int rg  = tid >> 5;                // 0..7 ; r = rg + 8*it, it<9
    const float2* r0p = (const float2*)&R[(2 * rg + 0) * STR + cc2];
    const float2* r1p = (const float2*)&R[(2 * rg + 1) * STR + cc2];
    float2* sp = (float2*)&S[rg * STR + cc2];
#pragma unroll
    for (int it = 0; it < 9; ++it) {
      float2 l0 = *r0p, l1 = *r1p;
      float2 sv;
      sv.x = l0.x + wy * (l1.x - l0.x);
      sv.y = l0.y + wy * (l1.y - l0.y);
      *sp = sv;
      r0p += 8 * STR;        // +16 raw rows (in float2 units)
      r1p += 8 * STR;
      sp  += 4 * STR;        // +8 S rows    (in float2 units)
    }
    for (int k = tid; k < NROWS * 4; k += BLOCK) {      // cols 64..67
      int r = k >> 2;
      int c4 = 64 + (k & 3);
      float l0 = R[(2 * r + 0) * STR + c4];
      float l1 = R[(2 * r + 1) * STR + c4];
      S[r * STR + c4] = l0 + wy * (l1 - l0);
    }
  }
#else
  for (int k = tid; k < NROWS * NCA; k += BLOCK) {
    int r  = k / NCA;
    int cc = k - r * NCA;
    int g  = min(x_al + cc, LUT_W - 1);
    const float* rowb = pb + (size_t)r * LUT_H * LUT_W;
    float l0 = rowb[(size_t)y0 * LUT_W + (size_t)g];
    float l1 = rowb[(size_t)y1 * LUT_W + (size_t)g];
    S[r * STR + cc] = l0 + wy * (l1 - l0);
  }
#endif
  __syncthreads();

  // =========== Per-pixel phase ===========
  float gxv = fmaf((float)w, sx, -1.0f);
  float ix = (gxv + 1.0f) * 0.5f * (float)(LUT_W - 1);
  ix = fminf(fmaxf(ix, 0.0f), (float)(LUT_W - 1));
  float x0f = floorf(ix);
  int x0 = (int)x0f;
  float wx = ix - x0f;
  int cc0 = x0 - x_al;
  cc0 = min(max(cc0, 0), NCA - 2);      // guard (never triggers in-range)

  float xv[3];
  xv[0] = x[pix];
  xv[1] = x[pix + cstride];
  xv[2] = x[pix + 2 * cstride];

  float acc[3] = {0.0f, 0.0f, 0.0f};
#pragma unroll
  for (int i = 0; i < 3; ++i) {
    float gz = xv[i];
    float iz = (gz + 1.0f) * 0.5f * (float)(LUT_D - 1);
    iz = fminf(fmaxf(iz, 0.0f), (float)(LUT_D - 1));
    float z0f = floorf(iz);
    int z0 = (int)z0f;
    int z1 = min(z0 + 1, LUT_D - 1);
    float wz = iz - z0f;
#pragma unroll
    for (int c = 0; c < 3; ++c) {
      const float* s0 = &S[(c * 24 + i * 8 + z0) * STR + cc0];
      const float* s1 = &S[(c * 24 + i * 8 + z1) * STR + cc0];
      float a0 = s0[0], a1 = s0[1];     // contiguous pair -> ds_load_2addr
      float b0 = s1[0], b1 = s1[1];
      float va = a0 + wx * (a1 - a0);
      float vb = b0 + wx * (b1 - b0);
      acc[c] += va + wz * (vb - va);
    }
  }

  out[pix]               = TANHF(acc[0]);
  out[pix + cstride]     = TANHF(acc[1]);
  out[pix + 2 * cstride] = TANHF(acc[2]);
}

extern "C" void kernel_launch(void* const* d_in, const int* in_sizes, int n_in,
                              void* d_out, int out_size, void* d_ws, size_t ws_size,
                              hipStream_t stream) {
  (void)in_sizes; (void)n_in; (void)out_size; (void)d_ws; (void)ws_size;
  const float* x     = (const float*)d_in[0];   // [4,3,1024,1024] f32
  const float* param = (const float*)d_in[1];   // [4,72,256,256]  f32
  float* out = (float*)d_out;                   // [4,3,1024,1024] f32
  const int nblocks = NBATCH * IMG_H * (IMG_W / BLOCK);  // 16384
  lut3d_tanh_kernel<<<nblocks, BLOCK, 0, stream>>>(x, param, out);
}